// Quantizer_9904194584677
// MI455X (gfx1250) — compile-verified
//
#include <hip/hip_runtime.h>
#include <hip/hip_bf16.h>

typedef __attribute__((ext_vector_type(2))) float v2f;
typedef __attribute__((ext_vector_type(4))) float v4f;
typedef __attribute__((ext_vector_type(8))) float v8f;

#define NTOT 65536   // 64*32*32
#define DIM  256
#define KCB  1024
#define BLOCK_ROWS 128   // 8 waves * 16 rows
#define KTILE 32         // codes staged per LDS tile
#define PAD 260          // row stride in floats (16B-aligned rows, bank-conflict padding)
#define NTILES (KCB / KTILE)

// ---------------------------------------------------------------- norms
__global__ void vq_norms(const float* __restrict__ cb, float* __restrict__ norms) {
    int k = blockIdx.x * blockDim.x + threadIdx.x;
    if (k >= KCB) return;
    const float* row = cb + (size_t)k * DIM;
    float s = 0.f;
#pragma unroll 4
    for (int c = 0; c < DIM; c += 4) {
        v4f q = *(const v4f*)(row + c);
        s += q.x * q.x + q.y * q.y + q.z * q.z + q.w * q.w;
    }
    norms[k] = s;
}

// Async-copy one 32x256 f32 codebook tile into LDS (padded rows), 8 B128 per thread.
// ISA 10.2: generic LDS pointers carry the LDS byte offset in addr[31:0].
__device__ __forceinline__ void async_tile(const float* __restrict__ gsrc,
                                           unsigned lds_base_bytes, int tid) {
#pragma unroll
    for (int i = 0; i < 8; ++i) {
        int e = tid + i * 256;          // float4 index, 2048 per tile
        int row = e >> 6, c4 = e & 63;  // 64 float4 per row
        unsigned laddr = lds_base_bytes + (unsigned)(row * PAD + c4 * 4) * 4u;
        const float* g = gsrc + row * DIM + c4 * 4;
        asm volatile("global_load_async_to_lds_b128 %0, %1, off"
                     :: "v"(laddr), "v"(g)
                     : "memory");
    }
}

// ---------------------------------------------------------------- main fused kernel
__global__ void __launch_bounds__(256)
vq_main(const float* __restrict__ z, const float* __restrict__ cb,
        const float* __restrict__ norms,
        float* __restrict__ out_q, float* __restrict__ out_idx,
        float* __restrict__ partials) {
    __shared__ float buf[2 * KTILE * PAD];   // double-buffered tiles; z staging reuses buf0/1
    __shared__ int   widx[8][16];
    __shared__ float werr[8];

    const int tid  = threadIdx.x;
    const int lane = tid & 31;
    const int w    = tid >> 5;           // wave 0..7
    const int col  = lane & 15;          // matrix column / M-row selector
    const int half = lane >> 4;          // 0: rows 0-7 / K 0-1 ; 1: rows 8-15 / K 2-3
    const int koff = half << 1;
    const int rowbase = blockIdx.x * BLOCK_ROWS + w * 16;   // wave's first global row

    // ---- Phase 1: stage z through LDS (coalesced) and build A-fragments ----
    // A layout (16x4 f32): lane l holds M=l%16, VGPR v holds K = 2*(l/16)+v
    v2f afr[64];
    for (int rr = 0; rr < 4; ++rr) {            // 4 rounds of 32 rows
        __syncthreads();
        const float* src = z + ((size_t)blockIdx.x * BLOCK_ROWS + rr * 32) * DIM;
#pragma unroll
        for (int i = 0; i < 8; ++i) {           // 2048 float4 per round / 256 threads
            int e = tid + i * 256;
            int row = e >> 6, c4 = e & 63;
            *(v4f*)(buf + row * PAD + c4 * 4) = *(const v4f*)(src + row * DIM + c4 * 4);
        }
        __syncthreads();
        if ((w >> 1) == rr) {
            int rloc = ((w & 1) << 4) + col;
            const float* ap = buf + rloc * PAD + koff;
#pragma unroll
            for (int d = 0; d < 64; ++d)
                afr[d] = *(const v2f*)(ap + d * 4);
        }
    }
    __syncthreads();   // all waves done reading staged z before tile 0 lands

    // ---- Phase 2: codebook sweep, async double-buffered tiles + fused argmin ----
    const unsigned lbase      = (unsigned)(size_t)buf;   // LDS byte offset of buf
    const unsigned tile_bytes = (unsigned)(KTILE * PAD * 4);

    async_tile(cb, lbase, tid);   // prefetch tile 0 -> buffer 0

    float mval[8];
    int   midx[8];
#pragma unroll
    for (int v = 0; v < 8; ++v) { mval[v] = 3.4e38f; midx[v] = 0; }

    for (int kt = 0; kt < NTILES; ++kt) {
        asm volatile("s_wait_asynccnt 0x0" ::: "memory");   // my tile-kt loads are in LDS
        __syncthreads();                                    // all loads landed; prev compute done
        if (kt + 1 < NTILES)                                // overlap next load with compute
            async_tile(cb + (size_t)(kt + 1) * KTILE * DIM,
                       lbase + (unsigned)((kt + 1) & 1) * tile_bytes, tid);

        const float* bufc = buf + (kt & 1) * (KTILE * PAD);
        // 4 independent accumulator chains: 2 subtiles x 2 D-halves hide WMMA RAW latency
        v8f acc0a = {}, acc0b = {}, acc1a = {}, acc1b = {};
        const float* bp0 = bufc + col * PAD + koff;          // subtile 0: codes col
        const float* bp1 = bufc + (16 + col) * PAD + koff;   // subtile 1: codes 16+col
#pragma unroll
        for (int d = 0; d < 32; ++d) {
            v2f b0a = *(const v2f*)(bp0 + d * 4);
            v2f b1a = *(const v2f*)(bp1 + d * 4);
            v2f b0b = *(const v2f*)(bp0 + (d + 32) * 4);
            v2f b1b = *(const v2f*)(bp1 + (d + 32) * 4);
            acc0a = __builtin_amdgcn_wmma_f32_16x16x4_f32(
                        false, afr[d],      false, b0a, (short)0, acc0a, false, false);
            acc1a = __builtin_amdgcn_wmma_f32_16x16x4_f32(
                        false, afr[d],      false, b1a, (short)0, acc1a, false, false);
            acc0b = __builtin_amdgcn_wmma_f32_16x16x4_f32(
                        false, afr[d + 32], false, b0b, (short)0, acc0b, false, false);
            acc1b = __builtin_amdgcn_wmma_f32_16x16x4_f32(
                        false, afr[d + 32], false, b1b, (short)0, acc1b, false, false);
        }
        // C layout: vgpr v, lane l -> row = v + 8*(l/16), code col = l%16
        int code0 = kt * KTILE + col;
        int code1 = code0 + 16;
        float n0 = norms[code0];
        float n1 = norms[code1];
#pragma unroll
        for (int v = 0; v < 8; ++v) {
            float s0 = n0 - 2.0f * (acc0a[v] + acc0b[v]);   // ||z||^2 dropped (row-constant)
            float s1 = n1 - 2.0f * (acc1a[v] + acc1b[v]);
            if (s0 < mval[v]) { mval[v] = s0; midx[v] = code0; }
            if (s1 < mval[v]) { mval[v] = s1; midx[v] = code1; }
        }
    }

    // ---- Phase 3: reduce argmin across the 16-lane column groups ----
#pragma unroll
    for (int m = 1; m < 16; m <<= 1) {
#pragma unroll
        for (int v = 0; v < 8; ++v) {
            float ov = __shfl_xor(mval[v], m, 32);
            int   oi = __shfl_xor(midx[v], m, 32);
            if (ov < mval[v] || (ov == mval[v] && oi < midx[v])) {
                mval[v] = ov; midx[v] = oi;
            }
        }
    }
    if (col == 0) {   // lanes 0 and 16 publish rows 0-7 / 8-15
#pragma unroll
        for (int v = 0; v < 8; ++v) {
            int r = half * 8 + v;
            out_idx[rowbase + r] = (float)midx[v];
            widx[w][r] = midx[v];
        }
    }

    // ---- Phase 4: gather winning codewords, write quantized, SSE partial ----
    float err = 0.f;
    for (int r = 0; r < 16; ++r) {
        int id   = widx[w][r];
        int grow = rowbase + r;
        const float* qp = cb + (size_t)id * DIM;
        const float* zp = z + (size_t)grow * DIM;
        float*       op = out_q + (size_t)grow * DIM;
#pragma unroll
        for (int i = 0; i < 2; ++i) {
            int c = lane * 8 + i * 4;
            v4f q  = *(const v4f*)(qp + c);
            v4f zz = *(const v4f*)(zp + c);
            v4f d  = q - zz;
            err += d.x * d.x + d.y * d.y + d.z * d.z + d.w * d.w;
            *(v4f*)(op + c) = q;
        }
    }
#pragma unroll
    for (int m = 1; m < 32; m <<= 1) err += __shfl_xor(err, m, 32);
    if (lane == 0) werr[w] = err;
    __syncthreads();
    if (tid == 0) {
        float s = 0.f;
#pragma unroll
        for (int i = 0; i < 8; ++i) s += werr[i];
        partials[blockIdx.x] = s;   // deterministic: no float atomics
    }
}

// ---------------------------------------------------------------- loss finalize
__global__ void vq_finalize(const float* __restrict__ partials, float* __restrict__ loss) {
    __shared__ float s[256];
    int t = threadIdx.x;
    s[t] = partials[t] + partials[t + 256];
    __syncthreads();
    for (int o = 128; o > 0; o >>= 1) {
        if (t < o) s[t] += s[t + o];
        __syncthreads();
    }
    // loss = embedding + 0.25*commitment = 1.25 * SSE / (N*D)
    if (t == 0) *loss = s[0] * (1.25f / ((float)NTOT * (float)DIM));
}

extern "C" void kernel_launch(void* const* d_in, const int* in_sizes, int n_in,
                              void* d_out, int out_size, void* d_ws, size_t ws_size,
                              hipStream_t stream) {
    const float* z  = (const float*)d_in[0];   // [64,32,32,256]
    const float* cb = (const float*)d_in[1];   // [1024,256]
    float* out      = (float*)d_out;           // [quantized | loss | idx]
    float* norms    = (float*)d_ws;            // 1024 floats
    float* partials = norms + KCB;             // 512 floats

    float* out_q    = out;
    float* out_loss = out + (size_t)NTOT * DIM;
    float* out_idx  = out_loss + 1;

    vq_norms<<<KCB / 256, 256, 0, stream>>>(cb, norms);
    vq_main<<<NTOT / BLOCK_ROWS, 256, 0, stream>>>(z, cb, norms, out_q, out_idx, partials);
    vq_finalize<<<1, 256, 0, stream>>>(partials, out_loss);
}